// GAT_3831110828332
// MI455X (gfx1250) — compile-verified
//
#include <hip/hip_runtime.h>
#include <hip/hip_bf16.h>

typedef __attribute__((ext_vector_type(2))) float v2f;
typedef __attribute__((ext_vector_type(8))) float v8f;
typedef __attribute__((ext_vector_type(4))) unsigned int u32x4;
typedef __attribute__((ext_vector_type(8))) int i32x8;
typedef __attribute__((ext_vector_type(4))) int i32x4;

// ---------------------------------------------------------------------------
// Exact fp32 GEMM via V_WMMA_F32_16X16X4_F32 with TDM-staged B panel in LDS.
// A: [M,K] rm, B: [K,Nout] rm, C: [M,Nout] rm.  Block = 8 waves; each block
// covers 128 rows x 64 cols.  Wave 0 issues one tensor_load_to_lds of the
// K x 64 B panel (64KB), waits TENSORcnt, then all waves read B frags from
// LDS. Each wave computes a 16x64 strip = 4 WMMA tiles sharing A fragments.
// ---------------------------------------------------------------------------
__global__ void gat_gemm_wmma_f32(const float* __restrict__ A,
                                  const float* __restrict__ B,
                                  float* __restrict__ Cmat,
                                  int M, int K, int Nout, int nGroups) {
    extern __shared__ float Bs[];                 // [K][64] panel, K <= 256
    const int lane = threadIdx.x & 31;
    const int wave = threadIdx.x >> 5;
    const int nGroup = blockIdx.x % nGroups;
    const int mGroup = blockIdx.x / nGroups;

    if (wave == 0) {
        // --- Tensor DMA descriptor (D#) for the K x 64 panel of B ---
        unsigned ldsOff = (unsigned)(size_t)(void*)Bs;   // low 32 bits = LDS byte offset
        unsigned long long ga =
            (unsigned long long)(size_t)(const void*)(B + (size_t)nGroup * 64);
        u32x4 g0;
        g0[0] = 1u;                                      // count=1, no gather
        g0[1] = ldsOff;                                  // lds_addr
        g0[2] = (unsigned)(ga & 0xFFFFFFFFull);          // global_addr[31:0]
        g0[3] = (unsigned)(ga >> 32) | (2u << 30);       // global_addr[56:32] | type=2
        i32x8 g1;
        g1[0] = (int)(2u << 16);                         // data_size=2 (4 bytes)
        g1[1] = (int)(((unsigned)Nout & 0xFFFFu) << 16); // tensor_dim0 lo
        g1[2] = (int)(((unsigned)Nout >> 16) |
                      (((unsigned)K & 0xFFFFu) << 16));  // tensor_dim0 hi | tensor_dim1 lo
        g1[3] = (int)(((unsigned)K >> 16) | (64u << 16));// tensor_dim1 hi | tile_dim0=64
        g1[4] = (int)(unsigned)K;                        // tile_dim1=K, tile_dim2=0
        g1[5] = (int)(unsigned)Nout;                     // tensor_dim0_stride lo
        g1[6] = 0;                                       // stride hi | dim1_stride lo
        g1[7] = 0;
        i32x4 gz = {0, 0, 0, 0};
#if defined(__clang_major__) && __clang_major__ >= 23
        i32x8 gz8 = {0, 0, 0, 0, 0, 0, 0, 0};
        __builtin_amdgcn_tensor_load_to_lds(g0, g1, gz, gz, gz8, 0);
#else
        __builtin_amdgcn_tensor_load_to_lds(g0, g1, gz, gz, 0);
#endif
        __builtin_amdgcn_s_wait_tensorcnt(0);
    }
    __syncthreads();

    const int tm = mGroup * 8 + wave;                    // 16-row tile index
    if (tm * 16 >= M) return;                            // wave-uniform, after barrier
    const int l15 = lane & 15;
    const int khalf = (lane >> 4) << 1;                  // 0 or 2

    int mrow = tm * 16 + l15; if (mrow >= M) mrow = M - 1;   // clamp (loads only)
    const float* Arow = A + (size_t)mrow * K;

    v8f acc[4];
#pragma unroll
    for (int nt = 0; nt < 4; ++nt) acc[nt] = (v8f){0.f,0.f,0.f,0.f,0.f,0.f,0.f,0.f};

    for (int k0 = 0; k0 < K; k0 += 4) {
        v2f a;
        a.x = Arow[k0 + khalf];
        a.y = Arow[k0 + khalf + 1];
        const float* b0p = &Bs[(k0 + khalf) * 64];
        const float* b1p = &Bs[(k0 + khalf + 1) * 64];
#pragma unroll
        for (int nt = 0; nt < 4; ++nt) {
            v2f b;
            b.x = b0p[nt * 16 + l15];
            b.y = b1p[nt * 16 + l15];
            acc[nt] = __builtin_amdgcn_wmma_f32_16x16x4_f32(
                    false, a, false, b, (short)0, acc[nt], false, false);
        }
    }

    const int mbase = tm * 16 + ((lane >> 4) << 3);
#pragma unroll
    for (int nt = 0; nt < 4; ++nt) {
        const int col = nGroup * 64 + nt * 16 + l15;
        if (col < Nout) {
#pragma unroll
            for (int r = 0; r < 8; ++r) {
                int row = mbase + r;
                if (row < M) Cmat[(size_t)row * Nout + col] = acc[nt][r];
            }
        }
    }
}

// Per-(node, head) attention logits: as[n,h] = <h[n,h,:], a_src[h,:]>, same for ad.
__global__ void gat_attn_logits(const float* __restrict__ h,
                                const float* __restrict__ a_src,
                                const float* __restrict__ a_dst,
                                float* __restrict__ as_, float* __restrict__ ad_,
                                int n, int H, int C) {
    int t = blockIdx.x * blockDim.x + threadIdx.x;
    if (t >= n * H) return;
    int node = t / H;
    int hh = t - node * H;
    const float* hp = h + (size_t)node * H * C + (size_t)hh * C;
    const float* sv = a_src + (size_t)hh * C;
    const float* dv = a_dst + (size_t)hh * C;
    float ssum = 0.f, dsum = 0.f;
    for (int cc = 0; cc < C; ++cc) {
        float v = hp[cc];
        ssum = fmaf(v, sv[cc], ssum);
        dsum = fmaf(v, dv[cc], dsum);
    }
    as_[t] = ssum;
    ad_[t] = dsum;
}

__global__ void gat_fill(float* __restrict__ p, float v, int n) {
    int t = blockIdx.x * blockDim.x + threadIdx.x;
    if (t < n) p[t] = v;
}

__device__ __forceinline__ void atomicMaxFloat(float* addr, float val) {
    // Works with init = -inf (0xFF800000): signed-max for val>=0, unsigned-min for val<0.
    if (val >= 0.f) atomicMax((int*)addr, __float_as_int(val));
    else            atomicMin((unsigned int*)addr, __float_as_uint(val));
}

// Segment max of leaky_relu(as[src]+ad[dst]) over incoming edges (incl. self loops).
__global__ void gat_edge_max(const long long* __restrict__ ei,
                             const float* __restrict__ as_,
                             const float* __restrict__ ad_,
                             float* __restrict__ emax,
                             int E, int n, int H) {
    int t = blockIdx.x * blockDim.x + threadIdx.x;
    int total = (E + n) * H;
    if (t >= total) return;
    int e = t / H;
    int hh = t - e * H;
    int s, d;
    if (e < E) { s = (int)ei[e]; d = (int)ei[E + e]; }
    else       { s = d = e - E; }
    float v = as_[s * H + hh] + ad_[d * H + hh];
    v = (v > 0.f) ? v : 0.2f * v;
    atomicMaxFloat(&emax[d * H + hh], v);
}

// Thread per (edge, head, channel): a wave covers 32 channels of one edge, so
// the h[src] gather is a coalesced 128B line and atomics issue lane-parallel.
__global__ void gat_edge_accum(const long long* __restrict__ ei,
                               const float* __restrict__ as_,
                               const float* __restrict__ ad_,
                               const float* __restrict__ emax,
                               const float* __restrict__ hfeat,
                               float* __restrict__ denom,
                               float* __restrict__ accum,
                               int E, int n, int H, int C) {
    long long t = (long long)blockIdx.x * blockDim.x + threadIdx.x;
    long long total = (long long)(E + n) * H * C;
    if (t >= total) return;
    int cc = (int)(t % C);
    long long eh = t / C;
    int hh = (int)(eh % H);
    int e  = (int)(eh / H);
    int s, d;
    if (e < E) { s = (int)ei[e]; d = (int)ei[E + e]; }
    else       { s = d = e - E; }
    float v = as_[s * H + hh] + ad_[d * H + hh];       // broadcast within wave
    v = (v > 0.f) ? v : 0.2f * v;
    float ex = __expf(v - emax[d * H + hh]);
    atomicAdd(&accum[(size_t)d * H * C + (size_t)hh * C + cc],
              ex * hfeat[(size_t)s * H * C + (size_t)hh * C + cc]);
    if (cc == 0) atomicAdd(&denom[d * H + hh], ex);
}

// act[n,h,c] = elu(accum[n,h,c] / denom[n,h] + bias[h*C+c])   (in place)
__global__ void gat_finalize_elu(float* __restrict__ acc,
                                 const float* __restrict__ denom,
                                 const float* __restrict__ bias,
                                 int n, int H, int C) {
    int t = blockIdx.x * blockDim.x + threadIdx.x;
    int HC = H * C;
    if (t >= n * HC) return;
    int node = t / HC;
    int rest = t - node * HC;
    int hh = rest / C;
    float v = acc[t] / denom[node * H + hh] + bias[rest];
    acc[t] = (v > 0.f) ? v : (__expf(v) - 1.0f);
}

// out[n,c] = out[n,c] / denom[n] + bias[c]   (layer 2: heads==1, mean==identity)
__global__ void gat_finalize_out(float* __restrict__ out,
                                 const float* __restrict__ denom,
                                 const float* __restrict__ bias,
                                 int n, int C) {
    int t = blockIdx.x * blockDim.x + threadIdx.x;
    if (t >= n * C) return;
    int node = t / C;
    int cc = t - node * C;
    out[t] = out[t] / denom[node] + bias[cc];
}

extern "C" void kernel_launch(void* const* d_in, const int* in_sizes, int n_in,
                              void* d_out, int out_size, void* d_ws, size_t ws_size,
                              hipStream_t stream) {
    const int IN_C = 256, H1 = 8, C1 = 32, HC1 = H1 * C1, OUT_C = 64;
    const float* x      = (const float*)d_in[0];
    const long long* ei = (const long long*)d_in[1];
    const float* W1     = (const float*)d_in[2];
    const float* a_src1 = (const float*)d_in[3];
    const float* a_dst1 = (const float*)d_in[4];
    const float* b1     = (const float*)d_in[5];
    const float* W2     = (const float*)d_in[6];
    const float* a_src2 = (const float*)d_in[7];
    const float* a_dst2 = (const float*)d_in[8];
    const float* b2     = (const float*)d_in[9];
    float* out = (float*)d_out;

    const int N = in_sizes[0] / IN_C;     // 50000
    const int E = in_sizes[1] / 2;        // 400000

    // Workspace carve-up (floats)
    float* ws   = (float*)d_ws;
    float* h1   = ws;                               // N*256
    float* act  = h1 + (size_t)N * HC1;             // N*256 (accum1, then ELU in place)
    float* as1  = act + (size_t)N * HC1;            // N*8
    float* ad1  = as1 + (size_t)N * H1;             // N*8
    float* emx1 = ad1 + (size_t)N * H1;             // N*8
    float* den1 = emx1 + (size_t)N * H1;            // N*8
    float* as2  = den1 + (size_t)N * H1;            // N
    float* ad2  = as2 + (size_t)N;                  // N
    float* emx2 = ad2 + (size_t)N;                  // N
    float* den2 = emx2 + (size_t)N;                 // N
    float* h2   = h1;                               // reuse: h1 dead after layer-1 accum

    const int TB = 256;
    auto cdiv = [](long long a, long long b) { return (int)((a + b - 1) / b); };

    const int mGroups = (N + 127) / 128;            // 128 rows per block (8 wave-tiles)
    const size_t ldsBytes = (size_t)IN_C * 64 * sizeof(float);   // 64KB B panel

    // ---------------- Layer 1 ----------------
    {
        int nGroups = HC1 / 64;                     // 4
        gat_gemm_wmma_f32<<<mGroups * nGroups, TB, ldsBytes, stream>>>(
            x, W1, h1, N, IN_C, HC1, nGroups);
    }
    gat_attn_logits<<<cdiv((long long)N * H1, TB), TB, 0, stream>>>(h1, a_src1, a_dst1,
                                                                    as1, ad1, N, H1, C1);
    gat_fill<<<cdiv((long long)N * H1, TB), TB, 0, stream>>>(emx1, -INFINITY, N * H1);
    gat_fill<<<cdiv((long long)N * H1, TB), TB, 0, stream>>>(den1, 0.f, N * H1);
    gat_fill<<<cdiv((long long)N * HC1, TB), TB, 0, stream>>>(act, 0.f, N * HC1);
    gat_edge_max<<<cdiv((long long)(E + N) * H1, TB), TB, 0, stream>>>(
        ei, as1, ad1, emx1, E, N, H1);
    gat_edge_accum<<<cdiv((long long)(E + N) * H1 * C1, TB), TB, 0, stream>>>(
        ei, as1, ad1, emx1, h1, den1, act, E, N, H1, C1);
    gat_finalize_elu<<<cdiv((long long)N * HC1, TB), TB, 0, stream>>>(act, den1, b1,
                                                                      N, H1, C1);

    // ---------------- Layer 2 (heads = 1, out = 64) ----------------
    {
        int nGroups = OUT_C / 64;                   // 1
        gat_gemm_wmma_f32<<<mGroups * nGroups, TB, ldsBytes, stream>>>(
            act, W2, h2, N, HC1, OUT_C, nGroups);
    }
    gat_attn_logits<<<cdiv((long long)N, TB), TB, 0, stream>>>(h2, a_src2, a_dst2,
                                                               as2, ad2, N, 1, OUT_C);
    gat_fill<<<cdiv((long long)N, TB), TB, 0, stream>>>(emx2, -INFINITY, N);
    gat_fill<<<cdiv((long long)N, TB), TB, 0, stream>>>(den2, 0.f, N);
    gat_fill<<<cdiv((long long)N * OUT_C, TB), TB, 0, stream>>>(out, 0.f, N * OUT_C);
    gat_edge_max<<<cdiv((long long)(E + N), TB), TB, 0, stream>>>(
        ei, as2, ad2, emx2, E, N, 1);
    gat_edge_accum<<<cdiv((long long)(E + N) * OUT_C, TB), TB, 0, stream>>>(
        ei, as2, ad2, emx2, h2, den2, out, E, N, 1, OUT_C);
    gat_finalize_out<<<cdiv((long long)N * OUT_C, TB), TB, 0, stream>>>(out, den2, b2,
                                                                        N, OUT_C);
}